// kNNClassifer_34445637714804
// MI455X (gfx1250) — compile-verified
//
#include <hip/hip_runtime.h>

// ---------------------------------------------------------------------------
// kNN classifier, fused f32-WMMA GEMM + streaming top-k  (gfx1250 / MI455X)
//   score[m,n] = r_sq[n] - 2 * (inputs @ data^T)[m,n]   (q_sq drops out of top-k)
// ---------------------------------------------------------------------------

typedef float v2f __attribute__((ext_vector_type(2)));
typedef float v8f __attribute__((ext_vector_type(8)));

#define M_TOTAL   2048
#define N_TOTAL   50000
#define D_DIM     512
#define K_TOP     10
#define NUM_CLS   10
#define NPARTS    4
#define CHUNK_COLS 128
#define CHUNKS_TOTAL     ((N_TOTAL + CHUNK_COLS - 1) / CHUNK_COLS)   // 391
#define CHUNKS_PER_PART  ((CHUNKS_TOTAL + NPARTS - 1) / NPARTS)      // 98

// LDS layout (floats), row stride padded to 516 to kill bank conflicts
#define LDA        516
#define A_OFF      0
#define B_OFF      (16 * LDA)                 // 8256
#define SCR_OFF    (B_OFF + CHUNK_COLS * LDA) // 74304
#define SMEM_FLOATS (SCR_OFF + 8 * 256)       // 76352 floats = 305,408 B (<320KB/WGP)

#define NEG_SENTINEL (-3.402823466e38f)
#define WAVES_PER_BLK 8
#define CANDS_PER_ROW (NPARTS * WAVES_PER_BLK * K_TOP)  // 320

// workspace byte offsets
#define WS_RSQ_OFF   0
#define WS_CS_OFF    200704                       // 256B-aligned, past 50000 f32
#define WS_CI_OFF    (WS_CS_OFF + (size_t)M_TOTAL * CANDS_PER_ROW * 4)

// ---------------------------------------------------------------------------
// Slow path: replace current min slot with (s,id); return the NEW min.
// Caller guarantees s > current min. Fully unrolled, register-resident.
// ---------------------------------------------------------------------------
__device__ __forceinline__ float topk_replace_min(float (&ts)[K_TOP], int (&ti)[K_TOP],
                                                  float s, int id) {
  float mn = ts[0];
  int mp = 0;
#pragma unroll
  for (int q = 1; q < K_TOP; ++q) {
    if (ts[q] < mn) { mn = ts[q]; mp = q; }
  }
#pragma unroll
  for (int q = 0; q < K_TOP; ++q) {
    ts[q] = (q == mp) ? s  : ts[q];
    ti[q] = (q == mp) ? id : ti[q];
  }
  float nm = ts[0];
#pragma unroll
  for (int q = 1; q < K_TOP; ++q) nm = fminf(nm, ts[q]);
  return nm;
}

// ---------------------------------------------------------------------------
// Kernel 1: r_sq[n] = sum(data[n,:]^2)  — one wave32 per row
// ---------------------------------------------------------------------------
__global__ __launch_bounds__(256) void knn_rsq_kernel(const float* __restrict__ data,
                                                      float* __restrict__ r_sq) {
  const int wave = blockIdx.x * (blockDim.x >> 5) + (threadIdx.x >> 5);
  const int l = threadIdx.x & 31;
  if (wave >= N_TOTAL) return;
  const float* rowp = data + (size_t)wave * D_DIM;
  float s = 0.0f;
#pragma unroll 4
  for (int d = l; d < D_DIM; d += 32) {
    const float v = rowp[d];
    s += v * v;
  }
#pragma unroll
  for (int off = 16; off > 0; off >>= 1) s += __shfl_xor(s, off, 32);
  if (l == 0) r_sq[wave] = s;
}

// ---------------------------------------------------------------------------
// Kernel 2: fused GEMM (v_wmma_f32_16x16x4_f32) + streaming per-row top-10
//   grid = (128 M-tiles, NPARTS N-partitions), block = 256 threads (8 waves)
// ---------------------------------------------------------------------------
__global__ __launch_bounds__(256, 1) void knn_gemm_topk_kernel(
    const float* __restrict__ inputs, const float* __restrict__ data,
    const float* __restrict__ r_sq,
    float* __restrict__ cand_s, int* __restrict__ cand_i) {
  extern __shared__ float smem[];

  const int tid  = threadIdx.x;
  const int w    = tid >> 5;        // wave id 0..7  -> N sub-tile
  const int l    = tid & 31;        // lane
  const int lr   = l & 15;
  const int hi   = l >> 4;          // half-wave selects K pair (A/B fragments)
  const int m0   = blockIdx.x * 16; // M tile base
  const int part = blockIdx.y;      // N partition

  // ---- stage A tile: 16 x 512 (reused for the whole N partition) ----
#pragma unroll
  for (int i = 0; i < 8; ++i) {
    const int f = tid + i * 256;          // float4 index, 0..2047
    const int row = f >> 7, c4 = f & 127;
    const float4 v = *reinterpret_cast<const float4*>(
        inputs + (size_t)(m0 + row) * D_DIM + c4 * 4);
    *reinterpret_cast<float4*>(&smem[A_OFF + row * LDA + c4 * 4]) = v;
  }

  float ts[K_TOP];
  int   ti[K_TOP];
#pragma unroll
  for (int q = 0; q < K_TOP; ++q) { ts[q] = NEG_SENTINEL; ti[q] = 0; }
  float thr = NEG_SENTINEL;   // cached min of ts[] (10th-best score)

  // Fragment base pointers: A[M=lr][K=kk+2*hi+{0,1}], B[N=w*16+lr][K=kk+2*hi+{0,1}]
  const float* aptr = &smem[A_OFF + lr * LDA + 2 * hi];
  const float* bptr = &smem[B_OFF + (w * 16 + lr) * LDA + 2 * hi];
  float* scr = &smem[SCR_OFF + w * 256];   // per-wave 16x16 score scratch

  const int c0 = part * CHUNKS_PER_PART;
  int c1 = c0 + CHUNKS_PER_PART;
  if (c1 > CHUNKS_TOTAL) c1 = CHUNKS_TOTAL;

  for (int c = c0; c < c1; ++c) {
    const int n0 = c * CHUNK_COLS;

    __syncthreads();  // previous chunk's B reads complete before restage
    // ---- stage B chunk: 128 x 512, fully coalesced float4 loads ----
#pragma unroll 4
    for (int i = 0; i < 64; ++i) {
      const int f = tid + i * 256;        // float4 index, 0..16383
      const int row = f >> 7, c4 = f & 127;
      int n = n0 + row;
      n = (n < N_TOTAL) ? n : (N_TOTAL - 1);   // clamp; OOB masked via score
      const float4 v = *reinterpret_cast<const float4*>(
          data + (size_t)n * D_DIM + c4 * 4);
      *reinterpret_cast<float4*>(&smem[B_OFF + row * LDA + c4 * 4]) = v;
    }
    __syncthreads();

    // ---- 16x16 cross tile via f32 WMMA, K = 512 in steps of 4 ----
    v8f acc = {};
#pragma unroll 4
    for (int kk = 0; kk < D_DIM; kk += 4) {
      const v2f a = *reinterpret_cast<const v2f*>(aptr + kk);
      const v2f b = *reinterpret_cast<const v2f*>(bptr + kk);
      acc = __builtin_amdgcn_wmma_f32_16x16x4_f32(
          false, a, false, b, (short)0, acc, false, false);
    }

    // ---- scores -> per-wave LDS scratch (C layout: row M=j+8*hi, col N=lr) ----
    const int ncol = n0 + w * 16 + lr;
    const bool valid = (ncol < N_TOTAL);
    const float rs = valid ? r_sq[ncol] : 0.0f;
#pragma unroll
    for (int j = 0; j < 8; ++j) {
      const float sc = valid ? (rs - 2.0f * acc[j]) : NEG_SENTINEL;
      scr[(j + 8 * hi) * 16 + lr] = sc;
    }
    __builtin_amdgcn_wave_barrier();  // keep store->load order within wave

    // ---- lanes 0..15 each own one row's running top-10.
    //      Fast path: single compare vs cached threshold; the full replace/
    //      re-min runs only on actual insertions (rare after warm-up). ----
    if (l < 16) {
      const float* rowp = scr + l * 16;
      const int nbase = n0 + w * 16;
#pragma unroll
      for (int v4 = 0; v4 < 4; ++v4) {
        const float4 sv = *reinterpret_cast<const float4*>(rowp + v4 * 4);
        const int ib = nbase + v4 * 4;
        if (sv.x > thr) thr = topk_replace_min(ts, ti, sv.x, ib + 0);
        if (sv.y > thr) thr = topk_replace_min(ts, ti, sv.y, ib + 1);
        if (sv.z > thr) thr = topk_replace_min(ts, ti, sv.z, ib + 2);
        if (sv.w > thr) thr = topk_replace_min(ts, ti, sv.w, ib + 3);
      }
    }
    __builtin_amdgcn_wave_barrier();
  }

  // ---- emit this wave's candidate lists ----
  if (l < 16) {
    const int m = m0 + l;
    const size_t base = (((size_t)m * NPARTS + part) * WAVES_PER_BLK + w) * K_TOP;
#pragma unroll
    for (int q = 0; q < K_TOP; ++q) {
      cand_s[base + q] = ts[q];
      cand_i[base + q] = ti[q];
    }
  }
}

// ---------------------------------------------------------------------------
// Kernel 3: per row merge 320 candidates -> top-10 -> labels -> mode -> pred
// ---------------------------------------------------------------------------
__global__ __launch_bounds__(256) void knn_final_kernel(const float* __restrict__ cand_s,
                                                        const int* __restrict__ cand_i,
                                                        float* __restrict__ out) {
  const int m = blockIdx.x * blockDim.x + threadIdx.x;
  if (m >= M_TOTAL) return;

  float ts[K_TOP];
  int   ti[K_TOP];
#pragma unroll
  for (int q = 0; q < K_TOP; ++q) { ts[q] = NEG_SENTINEL; ti[q] = 0; }
  float thr = NEG_SENTINEL;

  const float* s  = cand_s + (size_t)m * CANDS_PER_ROW;
  const int*   id = cand_i + (size_t)m * CANDS_PER_ROW;
  for (int j = 0; j < CANDS_PER_ROW; ++j) {
    const float sv = s[j];
    if (sv > thr) thr = topk_replace_min(ts, ti, sv, id[j]);
  }

  // label = floor(idx / (N/10)) == idx / 5000 for idx in [0, 50000)
  int lab[K_TOP];
#pragma unroll
  for (int q = 0; q < K_TOP; ++q) lab[q] = ti[q] / (N_TOTAL / NUM_CLS);

  // mode: max count, smallest class on ties (strict > keeps first)
  int bestc = 0, bestcnt = -1;
#pragma unroll
  for (int cls = 0; cls < NUM_CLS; ++cls) {
    int cnt = 0;
#pragma unroll
    for (int q = 0; q < K_TOP; ++q) cnt += (lab[q] == cls) ? 1 : 0;
    if (cnt > bestcnt) { bestcnt = cnt; bestc = cls; }
  }
  out[m] = (float)bestc;
}

// ---------------------------------------------------------------------------
// Host launcher
// ---------------------------------------------------------------------------
extern "C" void kernel_launch(void* const* d_in, const int* in_sizes, int n_in,
                              void* d_out, int out_size, void* d_ws, size_t ws_size,
                              hipStream_t stream) {
  (void)in_sizes; (void)n_in; (void)out_size; (void)ws_size;
  const float* inputs = (const float*)d_in[0];   // [2048, 512]
  const float* data   = (const float*)d_in[1];   // [50000, 512]
  // d_in[2] is k == 10 (compile-time K_TOP)
  float* out = (float*)d_out;                    // [2048] f32

  char* ws = (char*)d_ws;
  float* r_sq   = (float*)(ws + WS_RSQ_OFF);     // 50000 f32
  float* cand_s = (float*)(ws + WS_CS_OFF);      // 2048*320 f32
  int*   cand_i = (int*)  (ws + WS_CI_OFF);      // 2048*320 i32

  // 1) row norms of data: one wave per row, 8 waves/block
  knn_rsq_kernel<<<N_TOTAL / 8, 256, 0, stream>>>(data, r_sq);

  // 2) fused GEMM + streaming top-k
  dim3 grid2(M_TOTAL / 16, NPARTS);
  const size_t smem_bytes = (size_t)SMEM_FLOATS * sizeof(float);  // 305,408 B
  knn_gemm_topk_kernel<<<grid2, 256, smem_bytes, stream>>>(
      inputs, data, r_sq, cand_s, cand_i);

  // 3) merge candidates, label mode -> predictions
  knn_final_kernel<<<(M_TOTAL + 255) / 256, 256, 0, stream>>>(cand_s, cand_i, out);
}